// ScalarDotProductCriticNetwork_18614388261535
// MI455X (gfx1250) — compile-verified
//
#include <hip/hip_runtime.h>
#include <hip/hip_bf16.h>
#include <math.h>

typedef __attribute__((ext_vector_type(2))) float v2f;
typedef __attribute__((ext_vector_type(8))) float v8f;

#define BB 256
#define NN 32
#define OBS 128
#define ACT 8
#define DD 136
#define DK 128
#define DV 128
#define H1 64
#define OUTD 8
#define ROWS (BB*NN)          // 8192

__device__ __forceinline__ v8f wmma4(v2f a, v2f b, v8f c) {
    // V_WMMA_F32_16X16X4_F32 : D = A(16x4,f32) * B(4x16,f32) + C(16x16,f32)
    return __builtin_amdgcn_wmma_f32_16x16x4_f32(false, a, false, b, (short)0, c, false, false);
}

// ---------------------------------------------------------------------------
// Kernel 1: build obs_actions [ROWS,136] and obs_policy [ROWS,136]
// ---------------------------------------------------------------------------
__global__ void concat_kernel(const float* __restrict__ states,
                              const float* __restrict__ actions,
                              const float* __restrict__ policies,
                              float* __restrict__ oa, float* __restrict__ op) {
    int idx = blockIdx.x * blockDim.x + threadIdx.x;
    const int total = ROWS * DD;
    if (idx >= total) return;
    int r = idx / DD, c = idx - r * DD;
    if (c < OBS) {
        float s = states[r * OBS + c];
        oa[idx] = s;
        op[idx] = s;
    } else {
        oa[idx] = actions[r * ACT + (c - OBS)];
        op[idx] = policies[r * ACT + (c - OBS)];
    }
}

// ---------------------------------------------------------------------------
// Kernel 2: generic WMMA f32 GEMM  C[M,N] = act( (A - Asub) [M,K] @ W[K,N] )
// ACTF: 0 = identity, 1 = tanh.  SUB: 0 = plain A, 1 = A - Asub (for diff@W1).
// 4 waves per block, one 16x16 output tile per wave along M.
// Grid: (M/64, N/16), block 128. M%64==0, N%16==0, K%4==0 required.
// ---------------------------------------------------------------------------
template<int ACTF, int SUB>
__global__ void wmma_gemm_f32(const float* __restrict__ A,
                              const float* __restrict__ Asub,
                              const float* __restrict__ W,
                              float* __restrict__ C,
                              int M, int K, int N) {
    int wave = threadIdx.x >> 5;
    int lane = threadIdx.x & 31;
    int lo = lane & 15;
    int hi = lane >> 4;            // 0/1 half-wave
    int tm = (blockIdx.x * 4 + wave) * 16;   // row tile base
    int tn = blockIdx.y * 16;                // col tile base

    const float* arow = A + (size_t)(tm + lo) * K;
    const float* srow = SUB ? (Asub + (size_t)(tm + lo) * K) : nullptr;

    v8f acc = {};
    for (int kk = 0; kk < K; kk += 4) {
        v2f a, b;
        int ka = kk + 2 * hi;
        if (SUB) {
            a[0] = arow[ka]     - srow[ka];
            a[1] = arow[ka + 1] - srow[ka + 1];
        } else {
            a[0] = arow[ka];
            a[1] = arow[ka + 1];
        }
        b[0] = W[(size_t)ka * N + tn + lo];
        b[1] = W[(size_t)(ka + 1) * N + tn + lo];
        acc = wmma4(a, b, acc);
    }

    // D layout: VGPR v -> row (v + 8*hi), lane.lo -> col
    #pragma unroll
    for (int v = 0; v < 8; v++) {
        float val = acc[v];
        if (ACTF == 1) val = tanhf(val);
        C[(size_t)(tm + v + 8 * hi) * N + tn + lo] = val;
    }
}

// ---------------------------------------------------------------------------
// Kernel 3: per-batch attention. One block (128 thr = 4 waves) per batch b.
//   score = q[b] @ k[b]^T / sqrt(128)      (WMMA, 2x2 tiles, one per wave)
//   weight = softmax_rows(score)  -> LDS + global out
//   base[b] = weight @ av_act[b]           (WMMA, 2x8 tiles, 4 per wave)
// ---------------------------------------------------------------------------
__global__ void attention_kernel(const float* __restrict__ q,
                                 const float* __restrict__ k,
                                 const float* __restrict__ av_act,
                                 float* __restrict__ out_weight,
                                 float* __restrict__ base) {
    __shared__ float qs[NN * DK];    // 16 KB
    __shared__ float ks[NN * DK];    // 16 KB
    __shared__ float avs[NN * DV];   // 16 KB
    __shared__ float sw[NN * NN];    // 4 KB

    int b = blockIdx.x;
    int tid = threadIdx.x;
    int wave = tid >> 5;
    int lane = tid & 31;
    int lo = lane & 15;
    int hi = lane >> 4;

    // stage q, k, av_act for this batch into LDS (float4)
    {
        const float4* qg = (const float4*)(q      + (size_t)b * NN * DK);
        const float4* kg = (const float4*)(k      + (size_t)b * NN * DK);
        const float4* ag = (const float4*)(av_act + (size_t)b * NN * DV);
        float4* q4 = (float4*)qs;
        float4* k4 = (float4*)ks;
        float4* a4 = (float4*)avs;
        for (int i = tid; i < NN * DK / 4; i += blockDim.x) {
            q4[i] = qg[i];
            k4[i] = kg[i];
            a4[i] = ag[i];
        }
    }
    __syncthreads();

    // ---- score tiles: wave w owns tile (ti, tj), ti = w>>1, tj = w&1 ----
    {
        int ti = wave >> 1, tj = wave & 1;
        v8f acc = {};
        #pragma unroll 4
        for (int kk = 0; kk < DK; kk += 4) {
            v2f a, bb;
            int ka = kk + 2 * hi;
            a[0]  = qs[(ti * 16 + lo) * DK + ka];
            a[1]  = qs[(ti * 16 + lo) * DK + ka + 1];
            // B[krow][col] = k^T[krow][col] = k[col][krow]
            bb[0] = ks[(tj * 16 + lo) * DK + ka];
            bb[1] = ks[(tj * 16 + lo) * DK + ka + 1];
            acc = wmma4(a, bb, acc);
        }
        const float scale = 0.088388347648318447f;   // 1/sqrt(128)
        #pragma unroll
        for (int v = 0; v < 8; v++)
            sw[(ti * 16 + v + 8 * hi) * NN + tj * 16 + lo] = acc[v] * scale;
    }
    __syncthreads();

    // ---- softmax over j, one thread per row; write weight to global ----
    if (tid < NN) {
        float m = -3.402823466e+38f;
        for (int j = 0; j < NN; j++) m = fmaxf(m, sw[tid * NN + j]);
        float s = 0.f;
        for (int j = 0; j < NN; j++) {
            float e = __expf(sw[tid * NN + j] - m);
            sw[tid * NN + j] = e;
            s += e;
        }
        float inv = 1.f / s;
        for (int j = 0; j < NN; j++) {
            float wv = sw[tid * NN + j] * inv;
            sw[tid * NN + j] = wv;
            out_weight[(size_t)b * NN * NN + tid * NN + j] = wv;
        }
    }
    __syncthreads();

    // ---- base[b] = weight(32x32) @ av_act(32x128): 2x8 tiles, 4 per wave --
    for (int t = wave; t < 16; t += 4) {
        int ti = t >> 3;      // 0..1
        int tj = t & 7;       // 0..7
        v8f acc = {};
        #pragma unroll
        for (int kk = 0; kk < NN; kk += 4) {
            v2f a, bb;
            int ka = kk + 2 * hi;
            a[0]  = sw[(ti * 16 + lo) * NN + ka];
            a[1]  = sw[(ti * 16 + lo) * NN + ka + 1];
            bb[0] = avs[(size_t)ka * DV + tj * 16 + lo];
            bb[1] = avs[(size_t)(ka + 1) * DV + tj * 16 + lo];
            acc = wmma4(a, bb, acc);
        }
        #pragma unroll
        for (int v = 0; v < 8; v++)
            base[(size_t)b * NN * DV + (ti * 16 + v + 8 * hi) * DV + tj * 16 + lo] = acc[v];
    }
}

// ---------------------------------------------------------------------------
// Kernel 4: value[b,a,c,:] = leaky_relu(baseW1[b,a] + w[b,a,c]*diffW1[b,c]) @ W2
// One thread per (b,a,c).
// ---------------------------------------------------------------------------
__global__ void final_kernel(const float* __restrict__ baseW1,
                             const float* __restrict__ diffW1,
                             const float* __restrict__ weight,
                             const float* __restrict__ W2,
                             float* __restrict__ value) {
    int idx = blockIdx.x * blockDim.x + threadIdx.x;     // over B*N*N
    if (idx >= BB * NN * NN) return;
    int c = idx & (NN - 1);
    int a = (idx >> 5) & (NN - 1);
    int b = idx >> 10;

    const float* bw = baseW1 + (size_t)(b * NN + a) * H1;
    const float* dw = diffW1 + (size_t)(b * NN + c) * H1;
    float wv = weight[idx];

    float acc[OUTD];
    #pragma unroll
    for (int o = 0; o < OUTD; o++) acc[o] = 0.f;

    for (int t = 0; t < H1; t++) {
        float h = fmaf(wv, dw[t], bw[t]);
        h = h > 0.f ? h : 0.01f * h;                 // leaky_relu slope 0.01
        #pragma unroll
        for (int o = 0; o < OUTD; o++) acc[o] = fmaf(h, W2[t * OUTD + o], acc[o]);
    }
    float* out = value + (size_t)idx * OUTD;
    #pragma unroll
    for (int o = 0; o < OUTD; o++) out[o] = acc[o];
}

// ---------------------------------------------------------------------------
extern "C" void kernel_launch(void* const* d_in, const int* in_sizes, int n_in,
                              void* d_out, int out_size, void* d_ws, size_t ws_size,
                              hipStream_t stream) {
    // input order: states, policies, actions, Wk, Wq, Wv, W1, W2
    const float* states   = (const float*)d_in[0];
    const float* policies = (const float*)d_in[1];
    const float* actions  = (const float*)d_in[2];
    const float* Wk       = (const float*)d_in[3];
    const float* Wq       = (const float*)d_in[4];
    const float* Wv       = (const float*)d_in[5];
    const float* W1       = (const float*)d_in[6];
    const float* W2       = (const float*)d_in[7];

    float* out_value  = (float*)d_out;                       // [B,N,N,8]
    float* out_weight = out_value + (size_t)BB * NN * NN * OUTD; // [B,N,N]

    // workspace layout (floats)
    float* w  = (float*)d_ws;
    float* oa    = w;                       // ROWS*136
    float* op    = oa    + (size_t)ROWS * DD;
    float* q     = op    + (size_t)ROWS * DD;   // ROWS*128
    float* k     = q     + (size_t)ROWS * DK;
    float* avA   = k     + (size_t)ROWS * DK;   // tanh(obs_act@Wv)
    float* avP   = avA   + (size_t)ROWS * DV;   // tanh(obs_pol@Wv)
    float* base  = avP   + (size_t)ROWS * DV;   // weight@av_act
    float* bW1   = base  + (size_t)ROWS * DV;   // ROWS*64
    float* dW1   = bW1   + (size_t)ROWS * H1;

    // 1) concat
    {
        int total = ROWS * DD;
        concat_kernel<<<(total + 255) / 256, 256, 0, stream>>>(
            states, actions, policies, oa, op);
    }
    // 2) projections: q = states@Wq, k = states@Wk   [8192,128]@[128,128]
    wmma_gemm_f32<0, 0><<<dim3(ROWS / 64, DK / 16), 128, 0, stream>>>(
        states, nullptr, Wq, q, ROWS, OBS, DK);
    wmma_gemm_f32<0, 0><<<dim3(ROWS / 64, DK / 16), 128, 0, stream>>>(
        states, nullptr, Wk, k, ROWS, OBS, DK);
    // 3) av_act/av_pol = tanh(concat @ Wv)   [8192,136]@[136,128]
    wmma_gemm_f32<1, 0><<<dim3(ROWS / 64, DV / 16), 128, 0, stream>>>(
        oa, nullptr, Wv, avA, ROWS, DD, DV);
    wmma_gemm_f32<1, 0><<<dim3(ROWS / 64, DV / 16), 128, 0, stream>>>(
        op, nullptr, Wv, avP, ROWS, DD, DV);
    // 4) per-batch attention: score/softmax/weight + base = w@av_act
    attention_kernel<<<BB, 128, 0, stream>>>(q, k, avA, out_weight, base);
    // 5) baseW1 = base@W1 ; diffW1 = (avP-avA)@W1   [8192,128]@[128,64]
    wmma_gemm_f32<0, 0><<<dim3(ROWS / 64, H1 / 16), 128, 0, stream>>>(
        base, nullptr, W1, bW1, ROWS, DV, H1);
    wmma_gemm_f32<0, 1><<<dim3(ROWS / 64, H1 / 16), 128, 0, stream>>>(
        avP, avA, W1, dW1, ROWS, DV, H1);
    // 6) value tail
    {
        int total = BB * NN * NN;
        final_kernel<<<(total + 255) / 256, 256, 0, stream>>>(
            bW1, dW1, out_weight, W2, out_value);
    }
}